// TensoIRPhysicalRendering_58128087384968
// MI455X (gfx1250) — compile-verified
//
#include <hip/hip_runtime.h>
#include <hip/hip_bf16.h>
#include <stdint.h>

// ---------------- CDNA5 tuning constants ----------------
#define WPB    8      // waves per block (wave32): 256 threads
#define CHUNK  32     // lights per async chunk (one b128 async op = 512B)
#define NBUF   4      // async pipeline depth (stages of 512B per wave)
#define LMAX   512    // lights (reference fixes L=512)
#define PI_F    3.14159265358979323846f
#define FOURPI  12.5663706143591729539f

typedef int v4i __attribute__((ext_vector_type(4)));
typedef __attribute__((address_space(1))) v4i gv4i_t;   // global <4 x i32>
typedef __attribute__((address_space(3))) v4i lv4i_t;   // LDS    <4 x i32>

// ---- CDNA5 async global->LDS copy (16B per lane), tracked by ASYNCcnt ----
__device__ __forceinline__ void async_cp16(const float* g, float* l) {
#if __has_builtin(__builtin_amdgcn_global_load_async_to_lds_b128)
  __builtin_amdgcn_global_load_async_to_lds_b128((gv4i_t*)g, (lv4i_t*)l, 0, 0);
#else
  unsigned loff = (unsigned)(unsigned long long)(lv4i_t*)l;
  asm volatile("global_load_async_to_lds_b128 %0, %1, off"
               :: "v"(loff), "v"(g) : "memory");
#endif
}
__device__ __forceinline__ void wait_async2() {
  asm volatile("s_wait_asynccnt 0x2" ::: "memory");
}
__device__ __forceinline__ void wait_async1() {
  asm volatile("s_wait_asynccnt 0x1" ::: "memory");
}
__device__ __forceinline__ void wait_async0() {
  asm volatile("s_wait_asynccnt 0x0" ::: "memory");
}

// ---- fast fp32 transcendentals (single VALU trans ops) ----
__device__ __forceinline__ float rcp_f (float x){ return __builtin_amdgcn_rcpf(x);  }
__device__ __forceinline__ float rsq_f (float x){ return __builtin_amdgcn_rsqf(x);  }
__device__ __forceinline__ float exp2_f(float x){ return __builtin_amdgcn_exp2f(x); }
__device__ __forceinline__ float log2_f(float x){ return __builtin_amdgcn_logf(x);  }

__device__ __forceinline__ float lin2srgb(float x) {
  float lin = 12.92f * x;
  float e   = 1.055f * exp2_f(log2_f(fmaxf(x, 1e-8f)) * (1.0f / 2.4f)) - 0.055f;
  return (x <= 0.0031308f) ? lin : e;
}

// wave32 xor-tree reduction
__device__ __forceinline__ float wred(float v) {
  v += __shfl_xor(v, 16, 32);
  v += __shfl_xor(v,  8, 32);
  v += __shfl_xor(v,  4, 32);
  v += __shfl_xor(v,  2, 32);
  v += __shfl_xor(v,  1, 32);
  return v;
}

__global__ __launch_bounds__(WPB * 32)
void tensoir_render_kernel(const float* __restrict__ viewdirs,
                           const float* __restrict__ albedo,
                           const float* __restrict__ roughness,
                           const float* __restrict__ fresnel,
                           const float* __restrict__ normal,
                           const float* __restrict__ light_dirs,
                           const float* __restrict__ lawp,
                           const float* __restrict__ envp,
                           const float* __restrict__ visp,
                           const float* __restrict__ indp,
                           float* __restrict__ out,
                           int N, int L)
{
  // per-light constants, packed float4 SoA: one ds_load_b128 each
  __shared__ float4 s_l4[LMAX];   // {Lhat.x, Lhat.y, Lhat.z, light_area_weight}
  __shared__ float4 s_e4[LMAX];   // {env.r, env.g, env.b, 0}
  // per-wave NBUF-deep async stage: [0..31]=vis, [32..127]=ind (3 per light)
  __shared__ float s_stage[WPB][NBUF][CHUNK * 4];

  const int tid = threadIdx.x;

  // ---- block-cooperative light preprocessing ----
  for (int l = tid; l < L; l += blockDim.x) {
    float x = light_dirs[3 * l + 0];
    float y = light_dirs[3 * l + 1];
    float z = light_dirs[3 * l + 2];
    float inv = rsq_f(fmaxf(x * x + y * y + z * z, 1e-12f));
    s_l4[l] = make_float4(x * inv, y * inv, z * inv, lawp[l]);
    s_e4[l] = make_float4(envp[3 * l + 0], envp[3 * l + 1], envp[3 * l + 2], 0.f);
  }
  __syncthreads();

  const int wave = tid >> 5;
  const int lane = tid & 31;
  const int n = blockIdx.x * WPB + wave;
  if (n >= N) return;

  // ---- per-point precompute (lane-redundant broadcast loads) ----
  float vx = viewdirs[3 * n + 0], vy = viewdirs[3 * n + 1], vz = viewdirs[3 * n + 2];
  {
    float inv = rsq_f(fmaxf(vx * vx + vy * vy + vz * vz, 1e-12f));
    vx = -vx * inv; vy = -vy * inv; vz = -vz * inv;      // V = normalize(-viewdirs)
  }
  float n0x = normal[3 * n + 0], n0y = normal[3 * n + 1], n0z = normal[3 * n + 2];
  {
    float inv = rsq_f(fmaxf(n0x * n0x + n0y * n0y + n0z * n0z, 1e-12f));
    n0x *= inv; n0y *= inv; n0z *= inv;                  // N0 (unflipped, for cosine)
  }
  float NoV0 = vx * n0x + vy * n0y + vz * n0z;
  float sgn  = (NoV0 > 0.f) ? 1.f : ((NoV0 < 0.f) ? -1.f : 0.f);
  float nx = n0x * sgn, ny = n0y * sgn, nz = n0z * sgn;  // flipped N (for GGX)
  float NoV = fminf(fmaxf(vx * nx + vy * ny + vz * nz, 1e-6f), 1.f);

  float rgh    = fminf(fmaxf(roughness[n] * 0.9f + 0.09f, 0.f), 1.f);
  float alpha  = rgh * rgh;
  float alpha2 = alpha * alpha;
  float k      = (alpha + 2.f * rgh + 1.f) * 0.125f;
  float omk    = 1.f - k;
  float nom1   = NoV * omk + k;
  float a2m1   = alpha2 - 1.f;
  float c1     = FOURPI * nom1;                          // fold nom1 into 4*pi

  float fr = fresnel[3 * n + 0], fg = fresnel[3 * n + 1], fb = fresnel[3 * n + 2];
  float ofr = 1.f - fr, ofg = 1.f - fg, ofb = 1.f - fb;
  float ar = fminf(fmaxf(albedo[3 * n + 0], 0.f), 1.f) * (1.f / PI_F);
  float ag = fminf(fmaxf(albedo[3 * n + 1], 0.f), 1.f) * (1.f / PI_F);
  float ab = fminf(fmaxf(albedo[3 * n + 2], 0.f), 1.f) * (1.f / PI_F);

  // ---- async pipeline setup ----
  float* stg_base = &s_stage[wave][0][0];
  const size_t vbase = (size_t)n * (size_t)L;   // visibility row (floats)
  const size_t ibase = vbase * 3;               // indirect row (floats)
  const bool  isv = (lane < 8);
  const int   sub = isv ? lane * 4 : 32 + (lane - 8) * 4;

  float wi_r = 0.f, wi_g = 0.f, wi_b = 0.f;   // sum brdf*(vis*env)*w
  float wv_r = 0.f, wv_g = 0.f, wv_b = 0.f;   // sum brdf*env*w
  float id_r = 0.f, id_g = 0.f, id_b = 0.f;   // sum brdf*ind*w

  const int nchunk = L / CHUNK;

  auto issue = [&](int c) {
    const float* src = isv
        ? (visp + vbase + (size_t)c * CHUNK + lane * 4)
        : (indp + ibase + (size_t)c * CHUNK * 3 + (lane - 8) * 4);
    async_cp16(src, stg_base + (c & (NBUF - 1)) * (CHUNK * 4) + sub);
  };

  auto process = [&](int c) {
    const float* stg = stg_base + (c & (NBUF - 1)) * (CHUNK * 4);
    const int l = c * CHUNK + lane;
    float4 ld = s_l4[l];
    float4 ev = s_e4[l];
    float visv = stg[lane];
    float i0 = stg[32 + lane * 3 + 0];
    float i1 = stg[32 + lane * 3 + 1];
    float i2 = stg[32 + lane * 3 + 2];

    // cosine uses UNFLIPPED normal (reference semantics)
    float cosine = fmaxf(ld.x * n0x + ld.y * n0y + ld.z * n0z, 0.f);
    float m = (cosine > 1e-6f) ? 1.f : 0.f;
    float w = cosine * ld.w;
    visv *= m; i0 *= m; i1 *= m; i2 *= m;

    // H = normalize((L+V)/2); rsq(max(d^2,eps^2)) == rcp(max(sqrt(d^2),eps))
    float hx = (ld.x + vx) * 0.5f, hy = (ld.y + vy) * 0.5f, hz = (ld.z + vz) * 0.5f;
    float hinv = rsq_f(fmaxf(hx * hx + hy * hy + hz * hz, 1e-12f));
    hx *= hinv; hy *= hinv; hz *= hinv;

    float NoL = fminf(fmaxf(nx * ld.x + ny * ld.y + nz * ld.z, 1e-6f), 1.f);
    float NoH = fminf(fmaxf(nx * hx + ny * hy + nz * hz, 1e-6f), 1.f);
    float VoH = fminf(fmaxf(vx * hx + vy * hy + vz * hz, 1e-6f), 1.f);

    float FMi = (-5.55473f * VoH - 6.98316f) * VoH;
    float p2  = exp2_f(FMi);

    float t    = NoH * NoH;
    float nom0 = t * a2m1 + 1.f;
    float nom2 = NoL * omk + k;
    float nom  = fminf(fmaxf(c1 * nom0 * nom0 * nom2, 1e-6f), FOURPI);
    float rn   = rcp_f(nom) * alpha2;          // alpha2/nom

    float sr  = (fr + ofr * p2) * rn;
    float sg2 = (fg + ofg * p2) * rn;
    float sb  = (fb + ofb * p2) * rn;

    float br = ar + sr, bg = ag + sg2, bb = ab + sb;   // brdf
    float bwr = br * w, bwg = bg * w, bwb = bb * w;
    float vdr = visv * ev.x, vdg = visv * ev.y, vdb = visv * ev.z;

    wi_r += bwr * vdr;  wi_g += bwg * vdg;  wi_b += bwb * vdb;
    wv_r += bwr * ev.x; wv_g += bwg * ev.y; wv_b += bwb * ev.z;
    id_r += bwr * i0;   id_g += bwg * i1;   id_b += bwb * i2;
  };

  // prime NBUF-1 chunks
  for (int p = 0; p < NBUF - 1 && p < nchunk; ++p) issue(p);

  // steady state: issue chunk c+NBUF-1, wait for chunk c, process
  int steady = nchunk - (NBUF - 1);
  if (steady < 0) steady = 0;
  int c = 0;
  for (; c < steady; ++c) {
    issue(c + NBUF - 1);
    wait_async2();                // <=2 outstanding -> chunk c has landed
    process(c);
  }
  // drain: remaining outstanding = nchunk-1-c
  for (; c < nchunk; ++c) {
    int rem = nchunk - 1 - c;
    if (rem >= 2)      wait_async2();
    else if (rem == 1) wait_async1();
    else               wait_async0();
    process(c);
  }

  // ---- wave32 reduction of 9 accumulators ----
  wi_r = wred(wi_r); wi_g = wred(wi_g); wi_b = wred(wi_b);
  wv_r = wred(wv_r); wv_g = wred(wv_g); wv_b = wred(wv_b);
  id_r = wred(id_r); id_g = wred(id_g); id_b = wred(id_b);

  if (lane == 0) {
    const size_t N3 = (size_t)N * 3;
    const size_t b  = (size_t)n * 3;
    // rgb = srgb(clip(wo_indir_sum + indir_sum, 0, 1))
    float rr = fminf(fmaxf(wi_r + id_r, 0.f), 1.f);
    float rg = fminf(fmaxf(wi_g + id_g, 0.f), 1.f);
    float rb = fminf(fmaxf(wi_b + id_b, 0.f), 1.f);
    out[b + 0]          = lin2srgb(rr);
    out[b + 1]          = lin2srgb(rg);
    out[b + 2]          = lin2srgb(rb);
    out[N3 + b + 0]     = lin2srgb(wi_r);
    out[N3 + b + 1]     = lin2srgb(wi_g);
    out[N3 + b + 2]     = lin2srgb(wi_b);
    out[2 * N3 + b + 0] = lin2srgb(wv_r);
    out[2 * N3 + b + 1] = lin2srgb(wv_g);
    out[2 * N3 + b + 2] = lin2srgb(wv_b);
    out[3 * N3 + b + 0] = lin2srgb(id_r);
    out[3 * N3 + b + 1] = lin2srgb(id_g);
    out[3 * N3 + b + 2] = lin2srgb(id_b);
  }
}

extern "C" void kernel_launch(void* const* d_in, const int* in_sizes, int n_in,
                              void* d_out, int out_size, void* d_ws, size_t ws_size,
                              hipStream_t stream) {
  const float* viewdirs  = (const float*)d_in[0];
  const float* albedo    = (const float*)d_in[1];
  const float* roughness = (const float*)d_in[2];
  const float* fresnel   = (const float*)d_in[3];
  const float* normal    = (const float*)d_in[4];
  const float* light_dirs= (const float*)d_in[5];
  const float* law       = (const float*)d_in[6];
  const float* env       = (const float*)d_in[7];
  const float* vis       = (const float*)d_in[8];
  const float* ind       = (const float*)d_in[9];
  float* out = (float*)d_out;

  const int N = in_sizes[2];   // roughness is (N,1)
  const int L = in_sizes[6];   // light_area_weight is (L,)

  dim3 block(WPB * 32);
  dim3 grid((N + WPB - 1) / WPB);
  tensoir_render_kernel<<<grid, block, 0, stream>>>(
      viewdirs, albedo, roughness, fresnel, normal,
      light_dirs, law, env, vis, ind, out, N, L);
}